// FragmentsToExpression_25769803776509
// MI455X (gfx1250) — compile-verified
//
#include <hip/hip_runtime.h>

typedef __attribute__((ext_vector_type(2))) float v2f;
typedef __attribute__((ext_vector_type(8))) float v8f;

// Kernel 1: out[c*gene_n + g] = bias1[gene_ix[g]]
__global__ __launch_bounds__(256)
void fe_init_kernel(float* __restrict__ out, const float* __restrict__ bias,
                    const int* __restrict__ gene_ix, int gene_n, int n) {
  int i = blockIdx.x * blockDim.x + threadIdx.x;
  if (i < n) out[i] = bias[gene_ix[i % gene_n]];
}

// Kernel 2: per-fragment scalar dot via WMMA, sorted-key run merge, atomic scatter.
//
// Wave layout per 16-fragment tile:
//   lane = l, m = l&15, half = l>>4; both halves handle fragment (base + m).
//   k-chunk s covers emb dims cb..cb+3 with cb = 4s; lane-half picks dim pair
//   (cb + 2*half, cb + 2*half + 1) = (sin f_j, cos f_j) of coordinate d = cb>=20.
//   A (16x4 f32, 2 VGPRs/lane): VGPR0 = sin, VGPR1 = cos  (matches ISA A layout:
//     lanes 0-15 -> K=0,1 ; lanes 16-31 -> K=2,3 of row m).
//   B (4x16 f32): all columns identical = w[4s+k]; per-lane VGPR0/1 = w[cb], w[cb+1]
//     (rows 0,1 in lanes 0-15, rows 2,3 in lanes 16-31, per C-matrix striping).
//   10 accumulating V_WMMA_F32_16X16X4_F32 -> D[m,n] = dot(emb[frag m], w) for all n.
__global__ __launch_bounds__(256)
void fe_main_kernel(const float* __restrict__ coords, const int* __restrict__ cgix,
                    const float* __restrict__ freqs, const float* __restrict__ w,
                    float* __restrict__ out, int F) {
  const int lane = threadIdx.x & 31;
  const int m    = lane & 15;
  const int half = lane >> 4;
  const int waveId = blockIdx.x * (blockDim.x >> 5) + (threadIdx.x >> 5);
  constexpr int TILES = 8;                    // 128 consecutive fragments per wave
  const int fragBase = waveId * (16 * TILES);

  // Per-lane coefficient preload (uniform within half-wave, L1/K$ resident).
  float fr[10], wa[10], wb[10];
  int dsel[10];
#pragma unroll
  for (int s = 0; s < 10; ++s) {
    int cb = 4 * s + 2 * half;                // emb dim pair handled by this lane
    int d  = (cb >= 20) ? 1 : 0;              // which coordinate
    fr[s]   = freqs[cb - 20 * d];             // freqs[2j] == freqs[2j+1]
    wa[s]   = w[cb];
    wb[s]   = w[cb + 1];
    dsel[s] = d;
  }

  for (int t = 0; t < TILES; ++t) {
    const int frag  = fragBase + t * 16 + m;
    const bool valid = (frag < F);
    float x0 = 0.f, x1 = 0.f;
    int seg = -1;
    if (valid) {
      x0  = coords[2 * frag + 0];
      x1  = coords[2 * frag + 1];
      seg = cgix[frag];
    }

    v8f acc = {0.f, 0.f, 0.f, 0.f, 0.f, 0.f, 0.f, 0.f};
#pragma unroll
    for (int s = 0; s < 10; ++s) {
      float x  = dsel[s] ? x1 : x0;
      float ph = x * fr[s];
      v2f a, b;
      a.x = valid ? __sinf(ph) : 0.f;         // shift 0
      a.y = valid ? __cosf(ph) : 0.f;         // shift pi/2
      b.x = wa[s];
      b.y = wb[s];
      // D = A x B + C ; accumulate the 40-dim contraction in 10 steps of K=4
      acc = __builtin_amdgcn_wmma_f32_16x16x4_f32(
          /*neg_a=*/false, a, /*neg_b=*/false, b,
          /*c_mod=*/(short)0, acc, /*reuse_a=*/false, /*reuse_b=*/false);
    }

    // Diagonal extract: lane p holds D[?, p%16]; sel_p = acc[p & 7] covers
    // D[m,m] at lane m (m<8, VGPR m) and lane m+16 (m>=8, VGPR m-8).
    int l3 = lane & 7;
    float t01 = (l3 & 1) ? acc[1] : acc[0];
    float t23 = (l3 & 1) ? acc[3] : acc[2];
    float t45 = (l3 & 1) ? acc[5] : acc[4];
    float t67 = (l3 & 1) ? acc[7] : acc[6];
    float u03 = (l3 & 2) ? t23 : t01;
    float u47 = (l3 & 2) ? t67 : t45;
    float sel = (l3 & 4) ? u47 : u03;
    int src = (m < 8) ? m : (m + 16);
    float v = __shfl(sel, src, 32);           // ds_bpermute: lane m gets D[m,m]

    // Segmented inclusive scan over the 16 sorted keys (duplicated per half).
#pragma unroll
    for (int dlt = 1; dlt < 16; dlt <<= 1) {
      float vo = __shfl_up(v, dlt, 16);
      int   ko = __shfl_up(seg, dlt, 16);
      if ((m >= dlt) && (ko == seg)) v += vo;
    }
    int segn = __shfl_down(seg, 1, 16);
    bool tail = (m == 15) || (segn != seg);
    if ((half == 0) && tail && (seg >= 0)) {
      // Merged run total -> single HW float atomic (no return -> STOREcnt path)
      unsafeAtomicAdd(&out[seg], v);
    }
  }
}

extern "C" void kernel_launch(void* const* d_in, const int* in_sizes, int n_in,
                              void* d_out, int out_size, void* d_ws, size_t ws_size,
                              hipStream_t stream) {
  // Input order: coords, cellxgene_ix, cell_n, gene_n, gene_ix, freqs, shifts, weight1, bias1
  const float* coords  = (const float*)d_in[0];
  const int*   cgix    = (const int*)d_in[1];
  const int*   gene_ix = (const int*)d_in[4];
  const float* freqs   = (const float*)d_in[5];
  const float* w       = (const float*)d_in[7];
  const float* bias    = (const float*)d_in[8];
  float* out = (float*)d_out;

  const int F      = in_sizes[1];     // number of fragments
  const int gene_n = in_sizes[8];     // bias1 length
  const int n      = out_size;        // cell_n * gene_n

  fe_init_kernel<<<(n + 255) / 256, 256, 0, stream>>>(out, bias, gene_ix, gene_n, n);

  const int fragsPerBlock = (256 / 32) * 16 * 8;   // 8 waves x 128 frags = 1024
  int blocks = (F + fragsPerBlock - 1) / fragsPerBlock;
  if (blocks < 1) blocks = 1;
  fe_main_kernel<<<blocks, 256, 0, stream>>>(coords, cgix, freqs, w, out, F);
}